// BahdanauAttention_83476984365343
// MI455X (gfx1250) — compile-verified
//
#include <hip/hip_runtime.h>
#include <hip/hip_bf16.h>

// context[b,d] = sum_t keys[b,t,d]  (softmax over size-1 axis == 1, rest of the
// reference graph is dead).  Bandwidth-bound: 256 MiB read -> ~11.5us floor at
// 23.3 TB/s.  The column-sum is realized as WMMA F32 16x16x4 with A == ones so
// the accumulation runs on the matrix pipe (v_wmma_f32_16x16x4_f32).

typedef __attribute__((ext_vector_type(2))) float v2f;
typedef __attribute__((ext_vector_type(8))) float v8f;

#define B_   32
#define T_   4096
#define D_   512
#define TCH  8              // chunks of T per (b, d-tile)
#define TPC  (T_ / TCH)     // 512 rows per chunk
#define DT   (D_ / 16)      // 32 d-tiles of width 16

// Phase 1: one wave per (b, d-tile, t-chunk).  Sums keys[b, t0:t0+TPC, d0:d0+16]
// over t with A(16x4)=ones WMMAs; D rows are all identical column sums.
__global__ void __launch_bounds__(256)
bahdanau_sum_partial(const float* __restrict__ keys, float* __restrict__ ws) {
    const int wave = (blockIdx.x * blockDim.x + threadIdx.x) >> 5;
    const int lane = threadIdx.x & 31;

    const int b  = wave / (DT * TCH);
    const int r  = wave % (DT * TCH);
    const int dt = r / TCH;
    const int tc = r % TCH;

    const int d0 = dt * 16;
    const int t0 = tc * TPC;
    const int n  = lane & 15;          // column within d-tile
    const int kb = (lane >> 4) << 1;   // K-row pair base: 0 (lanes 0-15) / 2 (lanes 16-31)

    // B-operand layout (K x N = 4 x 16), mirroring the 32-bit A 16x4 layout:
    //   VGPR0: lanes 0-15 -> K=0, lanes 16-31 -> K=2
    //   VGPR1: lanes 0-15 -> K=1, lanes 16-31 -> K=3
    const float* p = keys + ((size_t)b * T_ + (size_t)(t0 + kb)) * D_ + d0 + n;

    v2f a;  a.x = 1.0f; a.y = 1.0f;    // ones matrix (all lanes, both K slots)
    v8f c0 = {};
    v8f c1 = {};

    // 8 rows of T per iteration, two independent accumulators to break the
    // WMMA D->C dependency chain.
    #pragma unroll 4
    for (int i = 0; i < TPC / 8; ++i) {
        v2f bm0, bm1;
        bm0.x = p[0];
        bm0.y = p[D_];
        bm1.x = p[4 * D_];
        bm1.y = p[5 * D_];
        c0 = __builtin_amdgcn_wmma_f32_16x16x4_f32(
                 false, a, false, bm0, (short)0, c0, false, false);
        c1 = __builtin_amdgcn_wmma_f32_16x16x4_f32(
                 false, a, false, bm1, (short)0, c1, false, false);
        p += 8 * D_;
    }

    // C/D layout: VGPR0 (element 0), lanes 0-15 hold M=0, N=lane -> the column
    // sums.  All D rows are identical (A rows identical), row 0 suffices.
    if (lane < 16) {
        ws[((size_t)(b * DT + dt) * TCH + tc) * 16 + n] = c0[0] + c1[0];
    }
}

// Phase 2: reduce the TCH partials per output element.  16384 outputs.
__global__ void __launch_bounds__(256)
bahdanau_reduce(const float* __restrict__ ws, float* __restrict__ out) {
    const int o = blockIdx.x * blockDim.x + threadIdx.x;   // 0 .. B_*D_-1
    const int b  = o / D_;
    const int d  = o % D_;
    const int dt = d >> 4;
    const int n  = d & 15;
    const float* p = ws + (size_t)(b * DT + dt) * TCH * 16 + n;
    float s = 0.0f;
    #pragma unroll
    for (int tc = 0; tc < TCH; ++tc) s += p[tc * 16];
    out[o] = s;
}

extern "C" void kernel_launch(void* const* d_in, const int* in_sizes, int n_in,
                              void* d_out, int out_size, void* d_ws, size_t ws_size,
                              hipStream_t stream) {
    // inputs: 0=query[B,D] 1=keys[B,T,D] 2=Ws 3=Wh 4=W  -- only keys is live.
    const float* keys = (const float*)d_in[1];
    float* out = (float*)d_out;                 // [B, D] = 16384 floats
    float* ws  = (float*)d_ws;                  // B*DT*TCH*16 = 131072 floats (512 KiB)

    const int waves   = B_ * DT * TCH;          // 8192 waves
    const int threads = 256;
    const int blocks1 = waves * 32 / threads;   // 1024
    bahdanau_sum_partial<<<blocks1, threads, 0, stream>>>(keys, ws);

    const int blocks2 = (B_ * D_) / threads;    // 64
    bahdanau_reduce<<<blocks2, threads, 0, stream>>>(ws, out);
}